// SlotAttention_62878321214014
// MI455X (gfx1250) — compile-verified
//
#include <hip/hip_runtime.h>

// ---------------------------------------------------------------------------
// CDNA5 (gfx1250) Slot Attention forward, fp32 end-to-end.
// Matrix ops: V_WMMA_F32_16X16X4_F32 (wave32, 16x16 C tile in 8 VGPRs).
// Tile staging: GLOBAL_LOAD_ASYNC_TO_LDS_B128 (ASYNCcnt) for verbatim copies.
// LDS tiles padded to 260 floats/row: 16B row alignment for B128 async writes
// and conflict-free A-column reads (260 mod 64 = 4 -> bank 4m+k distinct).
// ---------------------------------------------------------------------------

typedef __attribute__((ext_vector_type(2))) float v2f;
typedef __attribute__((ext_vector_type(8))) float v8f;

__device__ __forceinline__ v8f zero8() {
    v8f z;
#pragma unroll
    for (int i = 0; i < 8; ++i) z[i] = 0.0f;
    return z;
}

// D = A(16x4) * B(4x16) + C   (fp32 WMMA)
__device__ __forceinline__ v8f wmma4(v2f a, v2f b, v8f c) {
    return __builtin_amdgcn_wmma_f32_16x16x4_f32(
        /*neg_a=*/false, a, /*neg_b=*/false, b,
        /*c_mod=*/(short)0, c, /*reuse_a=*/false, /*reuse_b=*/false);
}

// Async copy of `nchunks` 16-byte chunks (64 float4 chunks per 256-float row)
// from global into an LDS tile with 260-float row stride, then wait.
// Uses the CDNA5 async-DMA path: no VGPR round trip, tracked by ASYNCcnt.
__device__ __forceinline__ void async_tile_load(float (*dst)[260],
                                                const float* __restrict__ src,
                                                int src_stride_f, int nchunks,
                                                int tid, int nthreads) {
    for (int chunk = tid; chunk < nchunks; chunk += nthreads) {
        const int r  = chunk >> 6;
        const int c4 = (chunk & 63) << 2;
        const uint64_t ga   = (uint64_t)(uintptr_t)(src + (size_t)r * src_stride_f + c4);
        // low 32 bits of a flat LDS address == wave-relative LDS byte offset
        const uint32_t ldsa = (uint32_t)(uintptr_t)(&dst[r][c4]);
        asm volatile("global_load_async_to_lds_b128 %0, %1, off"
                     :: "v"(ldsa), "v"(ga) : "memory");
    }
    asm volatile("s_wait_asynccnt 0x0" ::: "memory");
}

// LayerNorm of rows x 256 cols held in LDS; 8 threads per row (row = tid>>3),
// shfl_xor reduction inside 8-lane groups. Works for 128 thr (16 rows) and
// 256 thr (32 rows).
__device__ __forceinline__ void ln_rows(float (*A)[260],
                                        const float* __restrict__ g,
                                        const float* __restrict__ b,
                                        int tid) {
    const int row = tid >> 3;
    const int sub = tid & 7;
    float s = 0.0f, s2 = 0.0f;
    for (int c = sub; c < 256; c += 8) {
        float x = A[row][c];
        s += x; s2 += x * x;
    }
    s  += __shfl_xor(s, 1);  s2 += __shfl_xor(s2, 1);
    s  += __shfl_xor(s, 2);  s2 += __shfl_xor(s2, 2);
    s  += __shfl_xor(s, 4);  s2 += __shfl_xor(s2, 4);
    const float mean = s * (1.0f / 256.0f);
    const float var  = s2 * (1.0f / 256.0f) - mean * mean;
    const float rstd = rsqrtf(var + 1e-5f);
    for (int c = sub; c < 256; c += 8)
        A[row][c] = (A[row][c] - mean) * rstd * g[c] + b[c];
}

// ---------------------------------------------------------------------------
// Kernel 0: slots_init = mu + z * sigma * mu
// ---------------------------------------------------------------------------
__global__ void init_slots_kernel(const float* __restrict__ z,
                                  const float* __restrict__ w,
                                  const float* __restrict__ sigma,
                                  float* __restrict__ slots, int total) {
    int i = blockIdx.x * blockDim.x + threadIdx.x;
    if (i < total) {
        float mu = w[i % (11 * 256)];
        slots[i] = mu + z[i] * sigma[0] * mu;
    }
}

// ---------------------------------------------------------------------------
// Kernel 1: feats_ln = LN(features); k = feats_ln @ Wk; v = feats_ln @ Wv
// 32 rows per block (2 WMMA row tiles), 8 waves of 32. Wave w: matrix
// (k if w<4 else v), 64 cols, BOTH row tiles -> every B fragment feeds two
// WMMAs (1 vmem load per WMMA).
// ---------------------------------------------------------------------------
__global__ void kv_proj_kernel(const float* __restrict__ feats,
                               const float* __restrict__ lng,
                               const float* __restrict__ lnb,
                               const float* __restrict__ Wk,
                               const float* __restrict__ Wv,
                               float* __restrict__ kout,
                               float* __restrict__ vout) {
    __shared__ alignas(16) float A[32][260];
    const int row0 = blockIdx.x * 32;
    const int tid  = threadIdx.x;

    async_tile_load(A, feats + (size_t)row0 * 256, 256, 32 * 64, tid, 256);
    __syncthreads();
    ln_rows(A, lng, lnb, tid);
    __syncthreads();

    const int lane = tid & 31, wave = tid >> 5;
    const int half = lane >> 4, l16 = lane & 15;
    const float* Wm = (wave < 4) ? Wk : Wv;
    float* outm     = (wave < 4) ? kout : vout;
    const int cbase = (wave & 3) * 64;

    v8f acc0[4], acc1[4];
#pragma unroll
    for (int t = 0; t < 4; ++t) { acc0[t] = zero8(); acc1[t] = zero8(); }

    for (int k0 = 0; k0 < 256; k0 += 4) {
        v2f a0, a1;
        a0.x = A[l16][k0 + 2 * half];
        a0.y = A[l16][k0 + 2 * half + 1];
        a1.x = A[16 + l16][k0 + 2 * half];
        a1.y = A[16 + l16][k0 + 2 * half + 1];
#pragma unroll
        for (int t = 0; t < 4; ++t) {
            const int n = cbase + t * 16 + l16;
            v2f bf;
            bf.x = Wm[(size_t)(k0 + 2 * half) * 256 + n];
            bf.y = Wm[(size_t)(k0 + 2 * half + 1) * 256 + n];
            acc0[t] = wmma4(a0, bf, acc0[t]);
            acc1[t] = wmma4(a1, bf, acc1[t]);
        }
    }
#pragma unroll
    for (int t = 0; t < 4; ++t) {
#pragma unroll
        for (int r = 0; r < 8; ++r) {
            const int m   = r + 8 * half;
            const int col = cbase + t * 16 + l16;
            outm[(size_t)(row0 + m) * 256 + col]      = acc0[t][r];
            outm[(size_t)(row0 + 16 + m) * 256 + col] = acc1[t][r];
        }
    }
}

// ---------------------------------------------------------------------------
// Kernel 2 (generic): out[352,ncols] = [relu](LN(in[352,256]) @ W + bias)
// grid = (22, ncols/64), block = 128 (4 waves x 16-col tiles).
// ---------------------------------------------------------------------------
__global__ void ln_gemm_kernel(const float* __restrict__ in,
                               const float* __restrict__ g,
                               const float* __restrict__ bb,
                               const float* __restrict__ W, int ncols,
                               const float* __restrict__ bias, int do_relu,
                               float* __restrict__ out) {
    __shared__ alignas(16) float A[16][260];
    const int row0 = blockIdx.x * 16;
    const int tid  = threadIdx.x;

    async_tile_load(A, in + (size_t)row0 * 256, 256, 16 * 64, tid, 128);
    __syncthreads();
    ln_rows(A, g, bb, tid);
    __syncthreads();

    const int lane = tid & 31, wave = tid >> 5;
    const int half = lane >> 4, l16 = lane & 15;
    const int n = (blockIdx.y * 4 + wave) * 16 + l16;

    v8f acc = zero8();
    for (int k0 = 0; k0 < 256; k0 += 4) {
        v2f a, bf;
        a.x  = A[l16][k0 + 2 * half];
        a.y  = A[l16][k0 + 2 * half + 1];
        bf.x = W[(size_t)(k0 + 2 * half) * ncols + n];
        bf.y = W[(size_t)(k0 + 2 * half + 1) * ncols + n];
        acc = wmma4(a, bf, acc);
    }
    const float badd = bias ? bias[n] : 0.0f;
#pragma unroll
    for (int r = 0; r < 8; ++r) {
        const int m = r + 8 * half;
        float v = acc[r] + badd;
        if (do_relu) v = fmaxf(v, 0.0f);
        out[(size_t)(row0 + m) * ncols + n] = v;
    }
}

// ---------------------------------------------------------------------------
// Kernel 3: logits = (q @ k^T) * D^-0.5, softmax over SLOTS fully in-register
// (S=11 <= 16 rows of one WMMA tile). attn -> d_out region each iteration.
// grid = (N/64, B), block = 128; wave w owns j-tile (blockIdx.x*4+w).
// ---------------------------------------------------------------------------
__global__ void attn_kernel(const float* __restrict__ q,
                            const float* __restrict__ kmat,
                            float* __restrict__ attn_out) {
    __shared__ alignas(16) float Q[16][260];
    __shared__ alignas(16) float KT[4][16][260];
    const int b   = blockIdx.y;
    const int jt0 = blockIdx.x * 4;
    const int tid = threadIdx.x, lane = tid & 31, wave = tid >> 5;

    // rows 0..10 of q via async DMA; rows 11..15 zero-padded
    async_tile_load(Q, q + (size_t)b * 11 * 256, 256, 11 * 64, tid, 128);
    for (int idx = tid; idx < 5 * 256; idx += 128)
        Q[11 + (idx >> 8)][idx & 255] = 0.0f;
    // per-wave k tile (verbatim) via async DMA
    async_tile_load(KT[wave],
                    kmat + ((size_t)b * 4096 + (size_t)(jt0 + wave) * 16) * 256,
                    256, 16 * 64, lane, 32);
    __syncthreads();

    const int half = lane >> 4, l16 = lane & 15;
    v8f acc = zero8();
    for (int k0 = 0; k0 < 256; k0 += 4) {
        v2f a, bf;
        a.x  = Q[l16][k0 + 2 * half];
        a.y  = Q[l16][k0 + 2 * half + 1];
        bf.x = KT[wave][l16][k0 + 2 * half];
        bf.y = KT[wave][l16][k0 + 2 * half + 1];
        acc = wmma4(a, bf, acc);
    }

    // column j split across lane (rows 0-7) and lane^16 (rows 8-15, >=11 masked)
    const float scale = 0.0625f;  // 256^-0.5
    float e[8];
    float mx = -1e30f;
#pragma unroll
    for (int r = 0; r < 8; ++r) {
        const int m = r + 8 * half;
        const float x = (m < 11) ? acc[r] * scale : -1e30f;
        e[r] = x;
        mx = fmaxf(mx, x);
    }
    mx = fmaxf(mx, __shfl_xor(mx, 16));
    float se = 0.0f;
#pragma unroll
    for (int r = 0; r < 8; ++r) {
        const int m = r + 8 * half;
        const float v = (m < 11) ? __expf(e[r] - mx) : 0.0f;
        e[r] = v;
        se += v;
    }
    se += __shfl_xor(se, 16);
    const float rinv = 1.0f / se;

    const int j = (jt0 + wave) * 16 + l16;
#pragma unroll
    for (int r = 0; r < 8; ++r) {
        const int m = r + 8 * half;
        if (m < 11)
            attn_out[((size_t)b * 11 + m) * 4096 + j] = e[r] * rinv;
    }
}

// ---------------------------------------------------------------------------
// Kernel 4: deterministic row reduction: sums[row] = 1/(sum_j attn[row,j]+eps)
// ---------------------------------------------------------------------------
__global__ void rowsum_kernel(const float* __restrict__ attn,
                              float* __restrict__ sums) {
    __shared__ float red[256];
    const int row = blockIdx.x;  // 0..351
    const int tid = threadIdx.x;
    float s = 0.0f;
    for (int j = tid; j < 4096; j += 256) s += attn[(size_t)row * 4096 + j];
    red[tid] = s;
    __syncthreads();
    for (int off = 128; off > 0; off >>= 1) {
        if (tid < off) red[tid] += red[tid + off];
        __syncthreads();
    }
    if (tid == 0) sums[row] = 1.0f / (red[0] + 1e-6f);
}

// ---------------------------------------------------------------------------
// Kernel 5: updates = (attn * rowscale) @ v   [B,S,D], K = N = 4096.
// A chunks scaled on load (VALU transform -> normal path); v read coalesced.
// grid = (4, B), block = 128; wave col tile = blockIdx.x*4+wave.
// ---------------------------------------------------------------------------
__global__ void updates_kernel(const float* __restrict__ attn,
                               const float* __restrict__ rsums,
                               const float* __restrict__ v,
                               float* __restrict__ upd) {
    __shared__ alignas(16) float A[16][260];
    __shared__ float rsc[16];
    const int b = blockIdx.y, tid = threadIdx.x;
    const int lane = tid & 31, wave = tid >> 5;
    if (tid < 16) rsc[tid] = (tid < 11) ? rsums[b * 11 + tid] : 0.0f;

    const int half = lane >> 4, l16 = lane & 15;
    const int n = (blockIdx.x * 4 + wave) * 16 + l16;
    v8f acc = zero8();

    for (int kc = 0; kc < 4096; kc += 256) {
        __syncthreads();
        for (int idx = tid; idx < 16 * 256; idx += 128) {
            const int r = idx >> 8, c = idx & 255;
            A[r][c] = (r < 11)
                ? attn[((size_t)b * 11 + r) * 4096 + kc + c] * rsc[r]
                : 0.0f;
        }
        __syncthreads();
        for (int kk = 0; kk < 256; kk += 4) {
            v2f a, bf;
            a.x = A[l16][kk + 2 * half];
            a.y = A[l16][kk + 2 * half + 1];
            const size_t krow = (size_t)b * 4096 + kc + kk + 2 * half;
            bf.x = v[krow * 256 + n];
            bf.y = v[(krow + 1) * 256 + n];
            acc = wmma4(a, bf, acc);
        }
    }
#pragma unroll
    for (int r = 0; r < 8; ++r) {
        const int m = r + 8 * half;
        if (m < 11) upd[((size_t)b * 11 + m) * 256 + n] = acc[r];
    }
}

// ---------------------------------------------------------------------------
// Kernel 6: GRU cell. Six fused GEMM tiles per wave (gx/gh x r/z/n), gates
// in-register. grid = (22, 4), block = 128.
// ---------------------------------------------------------------------------
__global__ void gru_kernel(const float* __restrict__ upd,
                           const float* __restrict__ hprev,
                           const float* __restrict__ wih,
                           const float* __restrict__ whh,
                           const float* __restrict__ bih,
                           const float* __restrict__ bhh,
                           float* __restrict__ hnew) {
    __shared__ alignas(16) float X[16][260];
    __shared__ alignas(16) float H[16][260];
    const int row0 = blockIdx.x * 16;
    const int tid = threadIdx.x, lane = tid & 31, wave = tid >> 5;

    async_tile_load(X, upd + (size_t)row0 * 256, 256, 16 * 64, tid, 128);
    async_tile_load(H, hprev + (size_t)row0 * 256, 256, 16 * 64, tid, 128);
    __syncthreads();

    const int half = lane >> 4, l16 = lane & 15;
    const int c0 = (blockIdx.y * 4 + wave) * 16;

    v8f gx[3], gh[3];
#pragma unroll
    for (int g = 0; g < 3; ++g) { gx[g] = zero8(); gh[g] = zero8(); }

    for (int k0 = 0; k0 < 256; k0 += 4) {
        v2f ax, ah;
        ax.x = X[l16][k0 + 2 * half];
        ax.y = X[l16][k0 + 2 * half + 1];
        ah.x = H[l16][k0 + 2 * half];
        ah.y = H[l16][k0 + 2 * half + 1];
#pragma unroll
        for (int g = 0; g < 3; ++g) {
            const size_t wrow = (size_t)(g * 256 + c0 + l16) * 256;
            v2f bi, bh;
            bi.x = wih[wrow + k0 + 2 * half];
            bi.y = wih[wrow + k0 + 2 * half + 1];
            bh.x = whh[wrow + k0 + 2 * half];
            bh.y = whh[wrow + k0 + 2 * half + 1];
            gx[g] = wmma4(ax, bi, gx[g]);
            gh[g] = wmma4(ah, bh, gh[g]);
        }
    }

    const int col = c0 + l16;
    const float bir = bih[col], biz = bih[256 + col], bin = bih[512 + col];
    const float bhr = bhh[col], bhz = bhh[256 + col], bhn = bhh[512 + col];
#pragma unroll
    for (int r = 0; r < 8; ++r) {
        const int m = r + 8 * half;
        const float rg = 1.0f / (1.0f + __expf(-(gx[0][r] + bir + gh[0][r] + bhr)));
        const float zg = 1.0f / (1.0f + __expf(-(gx[1][r] + biz + gh[1][r] + bhz)));
        const float ng = tanhf(gx[2][r] + bin + rg * (gh[2][r] + bhn));
        const float h  = H[m][col];
        hnew[(size_t)(row0 + m) * 256 + col] = (1.0f - zg) * ng + zg * h;
    }
}

// ---------------------------------------------------------------------------
// Kernel 7: slots = resid + hidden @ w2 + b2   (K = 1024, LDS-chunked via
// async DMA). grid = (22, 4), block = 128.
// ---------------------------------------------------------------------------
__global__ void mlp2_kernel(const float* __restrict__ hidden,
                            const float* __restrict__ w2,
                            const float* __restrict__ b2,
                            const float* __restrict__ resid,
                            float* __restrict__ dst) {
    __shared__ alignas(16) float A[16][260];
    const int row0 = blockIdx.x * 16;
    const int tid = threadIdx.x, lane = tid & 31, wave = tid >> 5;
    const int half = lane >> 4, l16 = lane & 15;
    const int n = (blockIdx.y * 4 + wave) * 16 + l16;

    v8f acc = zero8();
    for (int kc = 0; kc < 1024; kc += 256) {
        __syncthreads();
        async_tile_load(A, hidden + (size_t)row0 * 1024 + kc, 1024, 16 * 64, tid, 128);
        __syncthreads();
        for (int kk = 0; kk < 256; kk += 4) {
            v2f a, bf;
            a.x  = A[l16][kk + 2 * half];
            a.y  = A[l16][kk + 2 * half + 1];
            bf.x = w2[(size_t)(kc + kk + 2 * half) * 256 + n];
            bf.y = w2[(size_t)(kc + kk + 2 * half + 1) * 256 + n];
            acc = wmma4(a, bf, acc);
        }
    }
#pragma unroll
    for (int r = 0; r < 8; ++r) {
        const int m = r + 8 * half;
        const size_t o = (size_t)(row0 + m) * 256 + n;
        dst[o] = resid[o] + acc[r] + b2[n];
    }
}

// ---------------------------------------------------------------------------
// Host launch
// ---------------------------------------------------------------------------
extern "C" void kernel_launch(void* const* d_in, const int* in_sizes, int n_in,
                              void* d_out, int out_size, void* d_ws, size_t ws_size,
                              hipStream_t stream) {
    (void)in_sizes; (void)n_in; (void)out_size; (void)ws_size;

    const float* features = (const float*)d_in[0];
    const float* sigma    = (const float*)d_in[1];
    const float* z        = (const float*)d_in[2];
    const float* slots_w  = (const float*)d_in[3];
    const float* ln_feat_g = (const float*)d_in[4];
    const float* ln_feat_b = (const float*)d_in[5];
    const float* ln_slot_g = (const float*)d_in[6];
    const float* ln_slot_b = (const float*)d_in[7];
    const float* ln_mlp_g  = (const float*)d_in[8];
    const float* ln_mlp_b  = (const float*)d_in[9];
    const float* Wk = (const float*)d_in[10];
    const float* Wv = (const float*)d_in[11];
    const float* Wq = (const float*)d_in[12];
    const float* gru_wih = (const float*)d_in[13];
    const float* gru_whh = (const float*)d_in[14];
    const float* gru_bih = (const float*)d_in[15];
    const float* gru_bhh = (const float*)d_in[16];
    const float* mlp_w1 = (const float*)d_in[17];
    const float* mlp_b1 = (const float*)d_in[18];
    const float* mlp_w2 = (const float*)d_in[19];
    const float* mlp_b2 = (const float*)d_in[20];

    float* out       = (float*)d_out;
    float* out_slots = out;            // [32,11,256] = 90112
    float* out_attn  = out + 90112;    // [32,11,4096]

    float* ws        = (float*)d_ws;
    float* kbuf      = ws;                              // 32*4096*256
    float* vbuf      = kbuf + (size_t)32 * 4096 * 256;
    float* slots_cur = vbuf + (size_t)32 * 4096 * 256;  // 352*256
    float* slots_tmp = slots_cur + 352 * 256;
    float* qbuf      = slots_tmp + 352 * 256;
    float* upd       = qbuf + 352 * 256;
    float* hidden    = upd + 352 * 256;                 // 352*1024
    float* rsums     = hidden + 352 * 1024;             // 352

    init_slots_kernel<<<(90112 + 255) / 256, 256, 0, stream>>>(
        z, slots_w, sigma, slots_cur, 90112);

    // LN(features) -> k, v   (B*N/32 = 4096 blocks of 32 rows)
    kv_proj_kernel<<<4096, 256, 0, stream>>>(
        features, ln_feat_g, ln_feat_b, Wk, Wv, kbuf, vbuf);

    for (int it = 0; it < 3; ++it) {
        ln_gemm_kernel<<<dim3(22, 4), 128, 0, stream>>>(
            slots_cur, ln_slot_g, ln_slot_b, Wq, 256, nullptr, 0, qbuf);

        attn_kernel<<<dim3(64, 32), 128, 0, stream>>>(qbuf, kbuf, out_attn);

        rowsum_kernel<<<352, 256, 0, stream>>>(out_attn, rsums);

        updates_kernel<<<dim3(4, 32), 128, 0, stream>>>(out_attn, rsums, vbuf, upd);

        gru_kernel<<<dim3(22, 4), 128, 0, stream>>>(
            upd, slots_cur, gru_wih, gru_whh, gru_bih, gru_bhh, slots_tmp);

        ln_gemm_kernel<<<dim3(22, 16), 128, 0, stream>>>(
            slots_tmp, ln_mlp_g, ln_mlp_b, mlp_w1, 1024, mlp_b1, 1, hidden);

        mlp2_kernel<<<dim3(22, 4), 128, 0, stream>>>(
            hidden, mlp_w2, mlp_b2, slots_tmp,
            (it == 2) ? out_slots : slots_cur);
    }
}